// GATModel_13804024889622
// MI455X (gfx1250) — compile-verified
//
#include <hip/hip_runtime.h>
#include <hip/hip_bf16.h>

typedef float v2f __attribute__((ext_vector_type(2)));
typedef float v8f __attribute__((ext_vector_type(8)));

#define NEG_SLOPE 0.2f

// ---------------------------------------------------------------------------
// Monotone order-preserving float<->uint key so segment-max can use
// atomicMax(u32) with a memset(0) sentinel (key 0 == -infinity-ish).
// ---------------------------------------------------------------------------
__device__ __forceinline__ unsigned fkey(float f) {
  unsigned u = __float_as_uint(f);
  return (u & 0x80000000u) ? ~u : (u | 0x80000000u);
}
__device__ __forceinline__ float fdecode(unsigned k) {
  return (k & 0x80000000u) ? __uint_as_float(k & 0x7FFFFFFFu)
                           : __uint_as_float(~k);
}

// Byte offset of a __shared__ object inside LDS (generic -> addrspace(3)).
__device__ __forceinline__ unsigned lds_offset_of(const void* p) {
  return (unsigned)(unsigned long long)
      (__attribute__((address_space(3))) const void*)p;
}

// ---------------------------------------------------------------------------
// WMMA fp32 GEMM: C[N,cols] = A[N,128] * Bt[cols,128]^T  (Bt pre-transposed).
// blockIdx.x -> 16-row strip; threadIdx.y -> 16-col tile (one wave per tile).
// A strip staged in LDS via GLOBAL_LOAD_ASYNC_TO_LDS_B128 (ASYNCcnt),
// then each wave runs 32x V_WMMA_F32_16X16X4_F32.
// ---------------------------------------------------------------------------
__global__ void __launch_bounds__(256)
wmma_gemm_f32(const float* __restrict__ A, const float* __restrict__ Bt,
              float* __restrict__ C, int N, int cols) {
  __shared__ float As[16 * 128];
  const int rowBase = blockIdx.x * 16;
  const int tid = threadIdx.y * 32 + threadIdx.x;
  const int nthr = blockDim.y * 32;

  // --- async stage of the 16x128 A strip: per-lane 16B global->LDS DMA ---
  for (int i = tid * 4; i < 16 * 128; i += nthr * 4) {
    int r = i >> 7, c = i & 127;
    int gr = rowBase + r; if (gr >= N) gr = N - 1;
    unsigned lds = lds_offset_of(&As[i]);
    unsigned long long ga =
        (unsigned long long)(const void*)&A[(size_t)gr * 128 + c];
    asm volatile("global_load_async_to_lds_b128 %0, %1, off"
                 :: "v"(lds), "v"(ga) : "memory");
  }
  asm volatile("s_wait_asynccnt 0" ::: "memory");
  __syncthreads();

  const int lane = threadIdx.x;          // 0..31 (wave32)
  const int colBase = threadIdx.y * 16;
  const int mRow = lane & 15;            // M (for A) / N (for B,C)
  const int kOff = (lane >> 4) * 2;      // lane-half selects K pair

  const float* __restrict__ Arow = &As[mRow * 128 + kOff];
  const float* __restrict__ Bcol = Bt + (size_t)(colBase + mRow) * 128 + kOff;

  v8f acc = {};
#pragma unroll 4
  for (int k = 0; k < 128; k += 4) {
    v2f a = *(const v2f*)&Arow[k];   // one ds_load_b64
    v2f b = *(const v2f*)&Bcol[k];   // one global_load_b64
    acc = __builtin_amdgcn_wmma_f32_16x16x4_f32(
        /*neg_a=*/false, a, /*neg_b=*/false, b,
        /*c_mod=*/(short)0, acc, /*reuse_a=*/false, /*reuse_b=*/false);
  }

  // C/D layout: VGPR r -> M=r (lanes 0-15), M=r+8 (lanes 16-31); N = lane&15
  const int mAdd = (lane >> 4) * 8;
  float* Cp = C + (size_t)(rowBase + mAdd) * cols + colBase + mRow;
  if (rowBase + 16 <= N) {               // uniform fast path: no guards
#pragma unroll
    for (int r = 0; r < 8; ++r) Cp[(size_t)r * cols] = acc[r];
  } else {
#pragma unroll
    for (int r = 0; r < 8; ++r)
      if (rowBase + mAdd + r < N) Cp[(size_t)r * cols] = acc[r];
  }
}

// ---------------------------------------------------------------------------
// Bt[c, k] = B[k, c] : pre-transpose weights so WMMA B fragments are b64 loads
// ---------------------------------------------------------------------------
__global__ void transpose_w(const float* __restrict__ B, float* __restrict__ Bt,
                            int K, int cols) {
  int i = blockIdx.x * blockDim.x + threadIdx.x;
  if (i >= K * cols) return;
  int c = i / K, kk = i - c * K;
  Bt[i] = B[(size_t)kk * cols + c];
}

// ---------------------------------------------------------------------------
// el[n,h] = sum_d feat[n,h,d]*al[h,d]; er likewise (one pass over feat).
// ---------------------------------------------------------------------------
__global__ void attn_terms(const float* __restrict__ feat,
                           const float* __restrict__ al,
                           const float* __restrict__ ar,
                           float* __restrict__ el, float* __restrict__ er,
                           int NH, int H, int D) {
  int i = blockIdx.x * blockDim.x + threadIdx.x;
  if (i >= NH) return;
  int h = i % H;
  const float4* f = (const float4*)(feat + (size_t)i * D);
  const float4* a = (const float4*)(al + h * D);
  const float4* r = (const float4*)(ar + h * D);
  float sl = 0.f, sr = 0.f;
  for (int d = 0; d < D / 4; ++d) {
    float4 v = f[d], x = a[d], y = r[d];
    sl += v.x * x.x + v.y * x.y + v.z * x.z + v.w * x.w;
    sr += v.x * y.x + v.y * y.y + v.z * y.z + v.w * y.w;
  }
  el[i] = sl; er[i] = sr;
}

// ---------------------------------------------------------------------------
// Pass 1: e = leaky_relu(el[src]+er[dst]); store; segment-max via key atomicMax.
// ---------------------------------------------------------------------------
__global__ void edge_logits(const int* __restrict__ src, const int* __restrict__ dst,
                            const float* __restrict__ el, const float* __restrict__ er,
                            float* __restrict__ ebuf, unsigned* __restrict__ mkey,
                            int EH, int H) {
  int i = blockIdx.x * blockDim.x + threadIdx.x;
  if (i >= EH) return;
  int e = i / H, h = i - e * H;
  int s = src[e], d = dst[e];
  float x = el[s * H + h] + er[d * H + h];
  float v = x > 0.f ? x : NEG_SLOPE * x;
  ebuf[i] = v;
  atomicMax(&mkey[d * H + h], fkey(v));
}

// ---------------------------------------------------------------------------
// Pass 2: ex = exp(e - m[dst]); store in-place; segment-sum denom via atomicAdd.
// ---------------------------------------------------------------------------
__global__ void edge_exp(const int* __restrict__ dst,
                         float* __restrict__ ebuf,
                         const unsigned* __restrict__ mkey,
                         float* __restrict__ denom, int EH, int H) {
  int i = blockIdx.x * blockDim.x + threadIdx.x;
  if (i >= EH) return;
  int e = i / H, h = i - e * H;
  int d = dst[e];
  float m = fdecode(mkey[d * H + h]);
  float ex = __expf(ebuf[i] - m);
  ebuf[i] = ex;
  atomicAdd(&denom[d * H + h], ex);
}

// ---------------------------------------------------------------------------
// Pass 3: out[dst,h,:] += (ex/denom[dst,h]) * feat[src,h,:]  (16-float chunks).
// Thread i covers (edge, head, chunk-of-16-within-D).
// ---------------------------------------------------------------------------
__global__ void edge_aggregate(const int* __restrict__ src, const int* __restrict__ dst,
                               const float* __restrict__ exbuf,
                               const float* __restrict__ denom,
                               const float* __restrict__ feat,
                               float* __restrict__ out,
                               int total, int H, int D, int CH) {
  int i = blockIdx.x * blockDim.x + threadIdx.x;
  if (i >= total) return;
  int c = i % CH;
  int eh = i / CH;
  int h = eh % H;
  int e = eh / H;
  int s = src[e], d = dst[e];
  float a = exbuf[e * H + h] / denom[d * H + h];
  const float* fs = feat + ((size_t)s * H + h) * D + c * 16;
  float* od = out + ((size_t)d * H + h) * D + c * 16;
#pragma unroll
  for (int j = 0; j < 16; j += 4) {
    float4 v = *(const float4*)&fs[j];
    atomicAdd(&od[j + 0], a * v.x);
    atomicAdd(&od[j + 1], a * v.y);
    atomicAdd(&od[j + 2], a * v.z);
    atomicAdd(&od[j + 3], a * v.w);
  }
}

__global__ void bias_relu(float* __restrict__ x, const float* __restrict__ b,
                          int total, int C) {
  int i = blockIdx.x * blockDim.x + threadIdx.x;
  if (i >= total) return;
  float v = x[i] + b[i % C];
  x[i] = v > 0.f ? v : 0.f;
}

// ---------------------------------------------------------------------------
// score[e] = sigmoid(dot64(h2[src], h2[dst]))  -- L2-resident gathers
// ---------------------------------------------------------------------------
__global__ void edge_score(const int* __restrict__ src, const int* __restrict__ dst,
                           const float* __restrict__ h2, float* __restrict__ out, int E) {
  int e = blockIdx.x * blockDim.x + threadIdx.x;
  if (e >= E) return;
  const float4* a = (const float4*)(h2 + (size_t)src[e] * 64);
  const float4* b = (const float4*)(h2 + (size_t)dst[e] * 64);
  float s = 0.f;
#pragma unroll
  for (int j = 0; j < 16; ++j) {
    float4 x = a[j], y = b[j];
    s += x.x * y.x + x.y * y.y + x.z * y.z + x.w * y.w;
  }
  out[e] = 1.f / (1.f + __expf(-s));
}

// ---------------------------------------------------------------------------
extern "C" void kernel_launch(void* const* d_in, const int* in_sizes, int n_in,
                              void* d_out, int out_size, void* d_ws, size_t ws_size,
                              hipStream_t stream) {
  const float* features = (const float*)d_in[0];
  const int*   src      = (const int*)d_in[1];
  const int*   dst      = (const int*)d_in[2];
  // d_in[3] = edge_type (unused by reference)
  const float* w1  = (const float*)d_in[4];
  const float* al1 = (const float*)d_in[5];
  const float* ar1 = (const float*)d_in[6];
  const float* b1  = (const float*)d_in[7];
  const float* w2  = (const float*)d_in[8];
  const float* al2 = (const float*)d_in[9];
  const float* ar2 = (const float*)d_in[10];
  const float* b2  = (const float*)d_in[11];
  float* out = (float*)d_out;

  const int N = in_sizes[0] / 128;
  const int E = in_sizes[1];
  const int H1 = 8, D1 = 16, HID = 128, FO = 64;

  // ---- workspace layout (bytes) ----
  char* ws = (char*)d_ws;
  size_t off = 0;
  auto alloc = [&](size_t bytes) { char* p = ws + off; off += (bytes + 255) & ~(size_t)255; return p; };
  float*    feat1  = (float*)alloc((size_t)N * HID * 4);   // reused as feat2 [N,64]
  float*    h1     = (float*)alloc((size_t)N * HID * 4);   // layer-1 aggregate / layer-2 input
  float*    h2     = (float*)alloc((size_t)N * FO * 4);
  float*    ebuf   = (float*)alloc((size_t)E * H1 * 4);    // layer2 uses first E floats
  float*    el1    = (float*)alloc((size_t)N * H1 * 4);
  float*    er1    = (float*)alloc((size_t)N * H1 * 4);
  unsigned* mkey1  = (unsigned*)alloc((size_t)N * H1 * 4);
  float*    den1   = (float*)alloc((size_t)N * H1 * 4);
  float*    el2    = (float*)alloc((size_t)N * 4);
  float*    er2    = (float*)alloc((size_t)N * 4);
  unsigned* mkey2  = (unsigned*)alloc((size_t)N * 4);
  float*    den2   = (float*)alloc((size_t)N * 4);
  float*    bt1    = (float*)alloc((size_t)HID * HID * 4); // W1^T [128,128]
  float*    bt2    = (float*)alloc((size_t)HID * FO * 4);  // W2^T [64,128]
  float*    feat2  = feat1;

  // ---- per-call re-init (graph-capture safe) ----
  hipMemsetAsync(h1, 0, (size_t)N * HID * 4, stream);
  hipMemsetAsync(h2, 0, (size_t)N * FO * 4, stream);
  hipMemsetAsync(mkey1, 0, (size_t)N * H1 * 4, stream);
  hipMemsetAsync(den1, 0, (size_t)N * H1 * 4, stream);
  hipMemsetAsync(mkey2, 0, (size_t)N * 4, stream);
  hipMemsetAsync(den2, 0, (size_t)N * 4, stream);

  const int rowTiles = (N + 15) / 16;

  // weight transposes (tiny)
  transpose_w<<<(HID * HID + 255) / 256, 256, 0, stream>>>(w1, bt1, HID, HID);
  transpose_w<<<(HID * FO + 255) / 256, 256, 0, stream>>>(w2, bt2, HID, FO);

  // ================= layer 1 =================
  wmma_gemm_f32<<<rowTiles, dim3(32, 8), 0, stream>>>(features, bt1, feat1, N, HID);
  {
    int NH = N * H1;
    attn_terms<<<(NH + 255) / 256, 256, 0, stream>>>(feat1, al1, ar1, el1, er1, NH, H1, D1);
  }
  {
    int EH = E * H1;
    edge_logits<<<(EH + 255) / 256, 256, 0, stream>>>(src, dst, el1, er1, ebuf, mkey1, EH, H1);
    edge_exp<<<(EH + 255) / 256, 256, 0, stream>>>(dst, ebuf, mkey1, den1, EH, H1);
    int tot = E * H1 * 1;  // D1 = 16 -> 1 chunk
    edge_aggregate<<<(tot + 255) / 256, 256, 0, stream>>>(src, dst, ebuf, den1, feat1, h1,
                                                          tot, H1, D1, 1);
  }
  bias_relu<<<((N * HID) + 255) / 256, 256, 0, stream>>>(h1, b1, N * HID, HID);

  // ================= layer 2 =================
  wmma_gemm_f32<<<rowTiles, dim3(32, 4), 0, stream>>>(h1, bt2, feat2, N, FO);
  attn_terms<<<(N + 255) / 256, 256, 0, stream>>>(feat2, al2, ar2, el2, er2, N, 1, FO);
  edge_logits<<<(E + 255) / 256, 256, 0, stream>>>(src, dst, el2, er2, ebuf, mkey2, E, 1);
  edge_exp<<<(E + 255) / 256, 256, 0, stream>>>(dst, ebuf, mkey2, den2, E, 1);
  {
    int tot = E * 1 * 4;  // FO = 64 -> 4 chunks of 16
    edge_aggregate<<<(tot + 255) / 256, 256, 0, stream>>>(src, dst, ebuf, den2, feat2, h2,
                                                          tot, 1, FO, 4);
  }
  bias_relu<<<((N * FO) + 255) / 256, 256, 0, stream>>>(h2, b2, N * FO, FO);

  // ================= edge predictor =================
  edge_score<<<(E + 255) / 256, 256, 0, stream>>>(src, dst, h2, out, E);
}